// InvariantPointAttention_44006234914908
// MI455X (gfx1250) — compile-verified
//
#include <hip/hip_runtime.h>

// ---------------------------------------------------------------------------
// Invariant Point Attention for MI455X (gfx1250), fp32 via V_WMMA_F32_16X16X4_F32.
// Memory-bound on streaming `pair` (512 MB) once. The pair stream is staged
// into LDS by the Tensor Data Mover (TENSOR_LOAD_TO_LDS, TENSORcnt double
// buffer); V tiles in the attention epilogue use GLOBAL_LOAD_ASYNC_TO_LDS_B128
// (ASYNCcnt double buffer). ~6.5 GFLOP of matrix math done with wave32 WMMA.
// ---------------------------------------------------------------------------

typedef __attribute__((ext_vector_type(2))) float v2f;
typedef __attribute__((ext_vector_type(8))) float v8f;
typedef unsigned int u32x4 __attribute__((ext_vector_type(4)));
typedef int i32x4 __attribute__((ext_vector_type(4)));
typedef int i32x8 __attribute__((ext_vector_type(8)));

constexpr int N   = 1024;
constexpr int C   = 384;
constexpr int DP  = 128;
constexpr int H   = 12;
constexpr int P   = 4;
constexpr int CH  = C / H;        // 32
constexpr int H3P = H * 3 * P;    // 144
constexpr float SCALING = 0.17677669529663687f;  // (C/H)^-0.5

__device__ __forceinline__ v8f wmma_f32(v2f a, v2f b, v8f c) {
  // 8 args: (neg_a, A, neg_b, B, c_mod, C, reuse_a, reuse_b)
  return __builtin_amdgcn_wmma_f32_16x16x4_f32(false, a, false, b, (short)0, c,
                                               false, false);
}

// Generic LDS pointer -> byte offset within wave's LDS (aperture low 32 bits).
__device__ __forceinline__ unsigned lds_off(const void* p) {
  return (unsigned)(unsigned long long)p;
}

// Async copy of 16 bytes global -> LDS (per active lane). Tracked by ASYNCcnt.
__device__ __forceinline__ void async_g2l_b128(unsigned lds_byte_off,
                                               const void* gsrc) {
  asm volatile("global_load_async_to_lds_b128 %0, %1, off"
               :: "v"(lds_byte_off), "v"((unsigned long long)gsrc)
               : "memory");
}

// TDM: DMA one 16x16-float 2-D tile (row stride `row_stride_elems` floats)
// from global into LDS (contiguous [16][16] layout). Tracked by TENSORcnt.
// D# bitfields per CDNA5 ISA. All operands are
// wave-uniform (SGPR-sourced descriptor).
__device__ __forceinline__ void tdm_load_16x16_f32(unsigned lds_byte,
                                                   const float* gsrc,
                                                   int row_stride_elems) {
  const unsigned long long ga = (unsigned long long)gsrc;
  u32x4 g0;
  g0[0] = 1u;                                   // count=1, user descriptor
  g0[1] = lds_byte;                             // lds_addr (bytes)
  g0[2] = (unsigned)ga;                         // global_addr[31:0]
  g0[3] = (unsigned)((ga >> 32) & 0x1FFFFFFull) // global_addr[56:32]
          | (2u << 30);                         // type = 2 ("image")
  i32x8 g1;
  g1[0] = 2 << 16;       // data_size = 4B; workgroup_mask = 0 (not in cluster)
  g1[1] = 16 << 16;      // tensor_dim0 = 16 (atomic_barrier_addr = 0)
  g1[2] = 16 << 16;      // tensor_dim1 = 16
  g1[3] = 16 << 16;      // tile_dim0 = 16
  g1[4] = 16;            // tile_dim1 = 16, tile_dim2 = 0 (2-D tile)
  g1[5] = row_stride_elems;  // tensor_dim0_stride[31:0] (data_size units)
  g1[6] = 0;             // stride[47:32] = 0; dim1_stride unused (2-D)
  g1[7] = 0;
  const i32x4 z4 = {0, 0, 0, 0};
#if defined(__clang_major__) && (__clang_major__ >= 23)
  const i32x8 z8 = {0, 0, 0, 0, 0, 0, 0, 0};
  __builtin_amdgcn_tensor_load_to_lds(g0, g1, z4, z4, z8, 0);
#else
  __builtin_amdgcn_tensor_load_to_lds(g0, g1, z4, z4, 0);
#endif
}

// ---------------------------------------------------------------------------
// Generic Y[M,Nout] = X[M,K] @ W[K,Nout] + bias. One wave per 16x16 tile.
// ---------------------------------------------------------------------------
__global__ __launch_bounds__(32) void k_gemm(const float* __restrict__ X,
                                             const float* __restrict__ W,
                                             const float* __restrict__ bias,
                                             float* __restrict__ Y,
                                             int K, int Nout) {
  const int ntn  = Nout / 16;
  const int tm   = blockIdx.x / ntn;
  const int tn   = blockIdx.x % ntn;
  const int lane = threadIdx.x;
  const int gr   = lane >> 4;     // lane group (0: K 0-1, 1: K 2-3)
  const int l16  = lane & 15;

  v8f acc = {};
  const float* xrow = X + (size_t)(tm * 16 + l16) * K;
  for (int k0 = 0; k0 < K; k0 += 4) {
    const int ka = k0 + gr * 2;
    v2f a; a[0] = xrow[ka];
    a[1] = xrow[ka + 1];
    v2f b; b[0] = W[(size_t)ka * Nout + tn * 16 + l16];
    b[1] = W[(size_t)(ka + 1) * Nout + tn * 16 + l16];
    acc = wmma_f32(a, b, acc);
  }
  const float bb = bias[tn * 16 + l16];
#pragma unroll
  for (int r = 0; r < 8; ++r) {
    const int row = tm * 16 + r + gr * 8;
    Y[(size_t)row * Nout + tn * 16 + l16] = acc[r] + bb;
  }
}

// ---------------------------------------------------------------------------
// Rotate local points to global frame: g[n][h][y*P+p] = sum_x R[n][y][x]*pts[..x..p]
// and s[n][h] = sum_{y,p} g^2. One thread per (n,h).
// ---------------------------------------------------------------------------
__global__ void k_points(const float* __restrict__ pts,  // [N][H*3*P]
                         const float* __restrict__ rot,  // [N][3][3]
                         float* __restrict__ g,          // [N][H][12]
                         float* __restrict__ s) {        // [N][H]
  const int idx = blockIdx.x * blockDim.x + threadIdx.x;
  if (idx >= N * H) return;
  const int n = idx / H, h = idx % H;
  const float* R  = rot + (size_t)n * 9;
  const float* lp = pts + (size_t)n * H3P + h * (3 * P);
  float* go = g + (size_t)idx * 12;
  float acc = 0.f;
#pragma unroll
  for (int y = 0; y < 3; ++y)
#pragma unroll
    for (int p = 0; p < P; ++p) {
      const float v = R[y * 3 + 0] * lp[0 * P + p] +
                      R[y * 3 + 1] * lp[1 * P + p] +
                      R[y * 3 + 2] * lp[2 * P + p];
      go[y * P + p] = v;
      acc += v * v;
    }
  s[idx] = acc;
}

// ---------------------------------------------------------------------------
// Full logits [H][N][N] for one 16x16 (n,m) tile per workgroup (12 waves).
// The 256x128 pair block is streamed from HBM exactly once: the TDM DMAs
// 16x16-float tiles into LDS (wave-private TENSORcnt double buffer) feeding
// the bias-projection WMMA. Then each wave computes its head's qk + point
// logits with WMMA.
// ---------------------------------------------------------------------------
__global__ __launch_bounds__(384) void k_logits(
    const float* __restrict__ pair,   // [N][N][DP]
    const float* __restrict__ Wp,     // [DP][H]
    const float* __restrict__ bp,     // [H]
    const float* __restrict__ q,      // [N][C] viewed [N][H][CH]
    const float* __restrict__ kk,     // [N][C]
    const float* __restrict__ qg,     // [N][H][12]
    const float* __restrict__ kg,     // [N][H][12]
    const float* __restrict__ sq,     // [N][H]
    const float* __restrict__ sk,     // [N][H]
    float* __restrict__ logits) {     // [H][N][N]
  // spair[rt][buf][m_local][d_in_chunk]: 16-float d-chunks, double buffered.
  __shared__ __align__(16) float spair[16][2][16][16];  // 32 KB
  __shared__ float sWp[DP][17];            // Wp padded (anti-bank-conflict)
  __shared__ float sbias[256][17];         // bias for 256 (n,m) pairs x heads

  const int tid  = threadIdx.x;
  const int wave = __builtin_amdgcn_readfirstlane(tid >> 5);
  const int lane = tid & 31;
  const int gr   = lane >> 4;
  const int l16  = lane & 15;
  const int n0   = blockIdx.x * 16;
  const int m0   = blockIdx.y * 16;

  for (int i = tid; i < DP * 16; i += blockDim.x) {
    const int d = i >> 4, hh = i & 15;
    sWp[d][hh] = (hh < H) ? Wp[d * H + hh] : 0.f;
  }
  __syncthreads();

  // ---- pair-bias GEMM. Row-tile rt covers rows (n0+rt, m0+0..15); wave w
  //      owns rt = w, w+12. Each 16-float d-chunk of the row-tile is one TDM
  //      descriptor (tile 16 m x 16 d, row stride DP); while chunk c is
  //      consumed by WMMA, chunk c+1 is in flight (TENSORcnt == 1).
  for (int rt = wave; rt < 16; rt += 12) {
    const float* growbase = pair + ((size_t)(n0 + rt) * N + m0) * DP;

    tdm_load_16x16_f32(lds_off(&spair[rt][0][0][0]), growbase, DP);

    v8f acc = {};
    for (int c = 0; c < 8; ++c) {
      const int buf = c & 1;
      if (c + 1 < 8) {  // issue next chunk into the other buffer
        tdm_load_16x16_f32(lds_off(&spair[rt][buf ^ 1][0][0]),
                           growbase + (c + 1) * 16, DP);
        asm volatile("s_wait_tensorcnt 0x1" ::: "memory");  // chunk c landed
      } else {
        asm volatile("s_wait_tensorcnt 0x0" ::: "memory");
      }
#pragma unroll
      for (int k0c = 0; k0c < 16; k0c += 4) {
        const int ka = k0c + gr * 2;
        const int d  = c * 16 + ka;
        v2f a; a[0] = spair[rt][buf][l16][ka];
        a[1] = spair[rt][buf][l16][ka + 1];
        v2f b; b[0] = sWp[d][l16];
        b[1] = sWp[d + 1][l16];
        acc = wmma_f32(a, b, acc);
      }
    }
#pragma unroll
    for (int r = 0; r < 8; ++r) sbias[rt * 16 + r + gr * 8][l16] = acc[r];
  }
  __syncthreads();

  // ---- per-head logits: wave w owns head w (blockDim = 12 waves exactly)
  const int h = wave;
  const float bph = bp[h];

  v8f cqk = {};
  const float* qrow = q + (size_t)(n0 + l16) * C + h * CH;
  const float* krow = kk + (size_t)(m0 + l16) * C + h * CH;
  for (int k0 = 0; k0 < CH; k0 += 4) {
    const int ka = k0 + gr * 2;
    v2f a; a[0] = qrow[ka]; a[1] = qrow[ka + 1];
    v2f b; b[0] = krow[ka]; b[1] = krow[ka + 1];
    cqk = wmma_f32(a, b, cqk);
  }

  v8f cpt = {};
  const float* qgr = qg + ((size_t)(n0 + l16) * H + h) * 12;
  const float* kgr = kg + ((size_t)(m0 + l16) * H + h) * 12;
  for (int k0 = 0; k0 < 12; k0 += 4) {
    const int ka = k0 + gr * 2;
    v2f a; a[0] = qgr[ka]; a[1] = qgr[ka + 1];
    v2f b; b[0] = kgr[ka]; b[1] = kgr[ka + 1];
    cpt = wmma_f32(a, b, cpt);
  }

  const float skm = sk[(size_t)(m0 + l16) * H + h];
  float* lo = logits + ((size_t)h * N + n0) * N + m0;
#pragma unroll
  for (int r = 0; r < 8; ++r) {
    const int ni  = r + gr * 8;
    const float sqn  = sq[(size_t)(n0 + ni) * H + h];
    const float bias = sbias[ni * 16 + l16][h] + bph;
    const float ptl  = -0.5f * (sqn + skm - 2.f * cpt[r]) * SCALING;
    lo[(size_t)ni * N + l16] = cqk[r] * SCALING + bias + ptl;
  }
}

// ---------------------------------------------------------------------------
// Softmax over m + attn@V for one (head, 16-query tile). One wave.
// Online max/sum via wave shuffles; exp tile staged in LDS for the WMMA
// A-fragment; V tiles double-buffered into LDS with async b128 copies.
// ---------------------------------------------------------------------------
__global__ __launch_bounds__(32) void k_softmax_av(
    const float* __restrict__ logits,  // [H][N][N]
    const float* __restrict__ v,       // [N][C] viewed [N][H][CH]
    float* __restrict__ outw) {        // [N][C] viewed [N][H][CH]
  __shared__ float pt[16][17];
  __shared__ __align__(16) float sv[2][16][32];  // V tiles, double buffered
  __shared__ float rmax[16], rsum[16];

  const int h    = blockIdx.x;
  const int tn   = blockIdx.y;
  const int lane = threadIdx.x;
  const int gr   = lane >> 4;
  const int l16  = lane & 15;
  const float* lbase = logits + ((size_t)h * N + tn * 16) * (size_t)N;

  // online row max + sum-of-exp
  for (int r = 0; r < 16; ++r) {
    const float* row = lbase + (size_t)r * N;
    float lm = -3.0e38f, ls = 0.f;
    for (int m = lane; m < N; m += 32) {
      const float x  = row[m];
      const float nm = fmaxf(lm, x);
      ls = ls * __expf(lm - nm) + __expf(x - nm);
      lm = nm;
    }
    for (int off = 16; off > 0; off >>= 1) {
      const float om = __shfl_xor(lm, off);
      const float os = __shfl_xor(ls, off);
      const float nm = fmaxf(lm, om);
      ls = ls * __expf(lm - nm) + os * __expf(om - nm);
      lm = nm;
    }
    if (lane == 0) { rmax[r] = lm; rsum[r] = ls; }
  }
  __syncthreads();

  // prologue: stage V tile 0 (4 async b128 per wave: 16 rows x 128 B)
  const int vm = lane >> 3, vs = (lane & 7) * 4;  // base transfer mapping
#pragma unroll
  for (int it = 0; it < 4; ++it) {
    const int m = vm + it * 4, sg = vs;
    async_g2l_b128(lds_off(&sv[0][m][sg]),
                   v + (size_t)m * C + h * CH + sg);
  }

  v8f acc0 = {}, acc1 = {};
  for (int mt = 0; mt < N / 16; ++mt) {
    const int buf = mt & 1;
    for (int i = lane; i < 256; i += 32) {
      const int r = i >> 4, m = i & 15;
      pt[r][m] = __expf(lbase[(size_t)r * N + mt * 16 + m] - rmax[r]);
    }
    __syncthreads();
    if (mt + 1 < N / 16) {  // stage next V tile into the other buffer
#pragma unroll
      for (int it = 0; it < 4; ++it) {
        const int m = vm + it * 4, sg = vs;
        async_g2l_b128(lds_off(&sv[buf ^ 1][m][sg]),
                       v + (size_t)((mt + 1) * 16 + m) * C + h * CH + sg);
      }
      asm volatile("s_wait_asynccnt 0x4" ::: "memory");  // tile mt landed
    } else {
      asm volatile("s_wait_asynccnt 0x0" ::: "memory");
    }
    for (int k0 = 0; k0 < 16; k0 += 4) {
      const int ka = k0 + gr * 2;
      v2f a; a[0] = pt[l16][ka]; a[1] = pt[l16][ka + 1];
      v2f b0; b0[0] = sv[buf][ka][l16];
      b0[1] = sv[buf][ka + 1][l16];
      acc0 = wmma_f32(a, b0, acc0);
      v2f b1; b1[0] = sv[buf][ka][16 + l16];
      b1[1] = sv[buf][ka + 1][16 + l16];
      acc1 = wmma_f32(a, b1, acc1);
    }
    __syncthreads();
  }

#pragma unroll
  for (int r = 0; r < 8; ++r) {
    const int row = r + gr * 8;
    const float inv = 1.f / rsum[row];
    const size_t o = (size_t)(tn * 16 + row) * C + h * CH;
    outw[o + l16]      = acc0[r] * inv;
    outw[o + 16 + l16] = acc1[r] * inv;
  }
}

// ---------------------------------------------------------------------------
// out = LayerNorm(single + proj) * gamma + beta. One block per row.
// ---------------------------------------------------------------------------
__global__ __launch_bounds__(128) void k_resid_ln(
    const float* __restrict__ single, const float* __restrict__ proj,
    const float* __restrict__ gamma, const float* __restrict__ beta,
    float* __restrict__ out) {
  __shared__ float ssum[128], ssq[128];
  const int n = blockIdx.x, tid = threadIdx.x;
  const float* xs = single + (size_t)n * C;
  const float* xp = proj + (size_t)n * C;
  float x[3], lsum = 0.f, lsq = 0.f;
#pragma unroll
  for (int i = 0; i < 3; ++i) {
    const float vv = xs[tid + i * 128] + xp[tid + i * 128];
    x[i] = vv; lsum += vv; lsq += vv * vv;
  }
  ssum[tid] = lsum; ssq[tid] = lsq;
  __syncthreads();
  for (int s = 64; s > 0; s >>= 1) {
    if (tid < s) { ssum[tid] += ssum[tid + s]; ssq[tid] += ssq[tid + s]; }
    __syncthreads();
  }
  const float mean = ssum[0] * (1.f / C);
  const float var  = ssq[0] * (1.f / C) - mean * mean;
  const float inv  = rsqrtf(var + 1e-5f);
#pragma unroll
  for (int i = 0; i < 3; ++i) {
    const int c = tid + i * 128;
    out[(size_t)n * C + c] = (x[i] - mean) * inv * gamma[c] + beta[c];
  }
}

// ---------------------------------------------------------------------------
extern "C" void kernel_launch(void* const* d_in, const int* in_sizes, int n_in,
                              void* d_out, int out_size, void* d_ws, size_t ws_size,
                              hipStream_t stream) {
  const float* single = (const float*)d_in[0];
  const float* pair   = (const float*)d_in[1];
  const float* rot    = (const float*)d_in[2];
  // d_in[3] = translations (unused by the reference)
  const float* Wq  = (const float*)d_in[4];  const float* bq  = (const float*)d_in[5];
  const float* Wk  = (const float*)d_in[6];  const float* bk  = (const float*)d_in[7];
  const float* Wv  = (const float*)d_in[8];  const float* bv  = (const float*)d_in[9];
  const float* Wp  = (const float*)d_in[10]; const float* bp  = (const float*)d_in[11];
  const float* Wpq = (const float*)d_in[12]; const float* bpq = (const float*)d_in[13];
  const float* Wpk = (const float*)d_in[14]; const float* bpk = (const float*)d_in[15];
  const float* Wo  = (const float*)d_in[16]; const float* bo  = (const float*)d_in[17];
  const float* gamma = (const float*)d_in[18]; const float* beta = (const float*)d_in[19];

  // Workspace layout (floats). Total ~15.2M floats (~58 MB).
  float* ws = (float*)d_ws;
  float* q    = ws;                  // N*C
  float* k    = q    + (size_t)N * C;
  float* v    = k    + (size_t)N * C;
  float* qpl  = v    + (size_t)N * C;      // N*H3P (local q points)
  float* kpl  = qpl  + (size_t)N * H3P;
  float* qg   = kpl  + (size_t)N * H3P;    // N*H*12 (global frame)
  float* kg   = qg   + (size_t)N * H * 12;
  float* sq   = kg   + (size_t)N * H * 12; // N*H
  float* sk   = sq   + (size_t)N * H;
  float* wtd  = sk   + (size_t)N * H;      // N*C  (attn @ v)
  float* proj = wtd  + (size_t)N * C;      // N*C  (out projection)
  float* lgt  = proj + (size_t)N * C;      // H*N*N logits/attn

  const dim3 b32(32);
  // scalar + point projections
  k_gemm<<<(N / 16) * (C / 16), b32, 0, stream>>>(single, Wq, bq, q, C, C);
  k_gemm<<<(N / 16) * (C / 16), b32, 0, stream>>>(single, Wk, bk, k, C, C);
  k_gemm<<<(N / 16) * (C / 16), b32, 0, stream>>>(single, Wv, bv, v, C, C);
  k_gemm<<<(N / 16) * (H3P / 16), b32, 0, stream>>>(single, Wpq, bpq, qpl, C, H3P);
  k_gemm<<<(N / 16) * (H3P / 16), b32, 0, stream>>>(single, Wpk, bpk, kpl, C, H3P);

  // rotate to global frame + squared norms
  k_points<<<(N * H + 127) / 128, 128, 0, stream>>>(qpl, rot, qg, sq);
  k_points<<<(N * H + 127) / 128, 128, 0, stream>>>(kpl, rot, kg, sk);

  // fused logits (qk + pair bias + point distances), pair streamed once via
  // TDM double buffering
  k_logits<<<dim3(N / 16, N / 16), 384, 0, stream>>>(pair, Wp, bp, q, k, qg, kg,
                                                     sq, sk, lgt);

  // softmax + attn @ v (async-to-LDS V tiles)
  k_softmax_av<<<dim3(H, N / 16), b32, 0, stream>>>(lgt, v, wtd);

  // output projection, then residual + LayerNorm
  k_gemm<<<(N / 16) * (C / 16), b32, 0, stream>>>(wtd, Wo, bo, proj, C, C);
  k_resid_ln<<<N, 128, 0, stream>>>(single, proj, gamma, beta, (float*)d_out);
}